// GCNGeneration_23450521436983
// MI455X (gfx1250) — compile-verified
//
#include <hip/hip_runtime.h>

// GCN layer: h = x@W^T (WMMA f32) ; scatter-add over edges ; bias+PReLU+ReLU.
// d_in: [0]=x f32[100000,128], [1]=edge_index i32[2,1.6M], [2]=edge_weight f32[1.6M],
//       [3]=W f32[128,128], [4]=bias f32[128], [5]=prelu_a f32[1]
// d_out: f32[100000,128].  d_ws: >= 51.2MB used for h.

typedef __attribute__((ext_vector_type(2))) float v2f;
typedef __attribute__((ext_vector_type(8))) float v8f;

#define N_NODES 100000
#define N_EDGES 1600000
#define D 128
#define XS_STRIDE 132  // 132 % 64 == 4 -> conflict-free LDS column access

// ---------------------------------------------------------------------------
// Kernel 0: zero-init the accumulator (d_out).
// ---------------------------------------------------------------------------
__global__ void gcn_zero(float4* __restrict__ out, int n4) {
  int i = blockIdx.x * blockDim.x + threadIdx.x;
  if (i < n4) out[i] = make_float4(0.f, 0.f, 0.f, 0.f);
}

// ---------------------------------------------------------------------------
// Kernel 1: h = x @ W^T via V_WMMA_F32_16X16X4_F32.
// One block (8 waves) per 16-row tile of x; wave w computes output columns
// [16w, 16w+16). K=128 -> 32 WMMA accumulate steps per wave.
//
// Fragment layouts (CDNA5 ISA 7.12.2):
//   A 16x4 f32 : lanes 0-15 hold M=lane, K={0,1}; lanes 16-31 hold M=lane-16, K={2,3}
//   B 4x16 f32 : lanes 0-15 hold N=lane, K={0,1}; lanes 16-31 hold N=lane-16, K={2,3}
//   C 16x16 f32: VGPR r, lanes 0-15 -> M=r, N=lane; lanes 16-31 -> M=r+8, N=lane-16
// B[k][n] = W[n][k] (since h = x @ W^T), so each lane streams row W[ncol][*].
// ---------------------------------------------------------------------------
__global__ __launch_bounds__(256) void gcn_gemm_wmma(const float* __restrict__ x,
                                                     const float* __restrict__ W,
                                                     float* __restrict__ h) {
  __shared__ float xs[16 * XS_STRIDE];

  const int tid = threadIdx.x;
  const int rowBase = blockIdx.x * 16;  // N_NODES % 16 == 0

  // Cooperative load: 16x128 f32 x-tile into padded LDS.
#pragma unroll
  for (int i = 0; i < 8; ++i) {
    int idx = tid + 256 * i;        // 0..2047
    int r = idx >> 7;               // row 0..15
    int c = idx & 127;              // col 0..127
    xs[r * XS_STRIDE + c] = x[(size_t)(rowBase + r) * D + c];
  }
  __syncthreads();

  const int wave = tid >> 5;
  const int lane = tid & 31;
  const int hi   = lane >> 4;       // 0: K base 0, 1: K base 2
  const int l15  = lane & 15;
  const int ncol = wave * 16 + l15; // output column owned by this lane

  const float* __restrict__ wrow = W + (size_t)ncol * D;  // B[k][ncol] = W[ncol][k]

  v8f c = {};
#pragma unroll 4
  for (int kk = 0; kk < D; kk += 4) {
    const int kb = kk + 2 * hi;
    v2f a, b;
    a.x = xs[l15 * XS_STRIDE + kb];
    a.y = xs[l15 * XS_STRIDE + kb + 1];
    b.x = wrow[kb];
    b.y = wrow[kb + 1];
    // 8-arg f32-family signature: (neg_a, A, neg_b, B, c_mod, C, reuse_a, reuse_b)
    c = __builtin_amdgcn_wmma_f32_16x16x4_f32(false, a, false, b, (short)0, c,
                                              false, false);
  }

#pragma unroll
  for (int r = 0; r < 8; ++r) {
    h[(size_t)(rowBase + r + 8 * hi) * D + ncol] = c[r];
  }
}

// ---------------------------------------------------------------------------
// Kernel 2: wave-per-edge scatter-add.  out[dst] += w_e * h[src].
// h and out both fit in the 192MB L2, so this is an L2-resident gather +
// L2-side f32 atomic stream.  Lane i owns float4 #i of the 128-wide row.
// ---------------------------------------------------------------------------
__global__ __launch_bounds__(256) void gcn_scatter(const float* __restrict__ h,
                                                   const int* __restrict__ ei,
                                                   const float* __restrict__ ew,
                                                   float* __restrict__ out) {
  const int gt = blockIdx.x * blockDim.x + threadIdx.x;
  const int e = gt >> 5;
  const int lane = gt & 31;
  if (e >= N_EDGES) return;

  const int src = ei[e];            // edge_index[0][e]
  const int dst = ei[N_EDGES + e];  // edge_index[1][e]
  const float w = ew[e];

  const float4* __restrict__ hv =
      (const float4*)(h + (size_t)src * D);
  float4 v = hv[lane];              // coalesced 512B row read, L2-hot

  float* __restrict__ o = out + (size_t)dst * D + lane * 4;
  atomicAdd(o + 0, v.x * w);
  atomicAdd(o + 1, v.y * w);
  atomicAdd(o + 2, v.z * w);
  atomicAdd(o + 3, v.w * w);
}

// ---------------------------------------------------------------------------
// Kernel 3: out = relu(prelu(out + bias)), in place, float4-vectorized.
// ---------------------------------------------------------------------------
__global__ void gcn_epilogue(float* __restrict__ out,
                             const float* __restrict__ bias,
                             const float* __restrict__ prelu_a) {
  const int n4 = N_NODES * D / 4;
  const int i = blockIdx.x * blockDim.x + threadIdx.x;
  if (i >= n4) return;

  const float a = prelu_a[0];
  float4* __restrict__ o4 = (float4*)out;
  const float4* __restrict__ b4 = (const float4*)bias;

  float4 v = o4[i];
  float4 b = b4[i & (D / 4 - 1)];
  v.x += b.x; v.y += b.y; v.z += b.z; v.w += b.w;

  v.x = fmaxf(v.x >= 0.f ? v.x : a * v.x, 0.f);
  v.y = fmaxf(v.y >= 0.f ? v.y : a * v.y, 0.f);
  v.z = fmaxf(v.z >= 0.f ? v.z : a * v.z, 0.f);
  v.w = fmaxf(v.w >= 0.f ? v.w : a * v.w, 0.f);

  o4[i] = v;
}

// ---------------------------------------------------------------------------
extern "C" void kernel_launch(void* const* d_in, const int* in_sizes, int n_in,
                              void* d_out, int out_size, void* d_ws, size_t ws_size,
                              hipStream_t stream) {
  const float* x    = (const float*)d_in[0];
  const int*   ei   = (const int*)d_in[1];
  const float* ew   = (const float*)d_in[2];
  const float* W    = (const float*)d_in[3];
  const float* bias = (const float*)d_in[4];
  const float* pa   = (const float*)d_in[5];
  float* out = (float*)d_out;
  float* h   = (float*)d_ws;  // 100000*128*4 = 51.2 MB scratch for h

  const int n4 = N_NODES * D / 4;  // 3.2M float4

  // 0) out = 0
  gcn_zero<<<(n4 + 255) / 256, 256, 0, stream>>>((float4*)out, n4);

  // 1) h = x @ W^T   (6250 blocks x 256 threads, WMMA f32)
  gcn_gemm_wmma<<<N_NODES / 16, 256, 0, stream>>>(x, W, h);

  // 2) scatter-add over edges: 1 wave per edge -> 1.6M waves, 8 waves/block
  gcn_scatter<<<N_EDGES / 8, 256, 0, stream>>>(h, ei, ew, out);

  // 3) bias + PReLU + ReLU in place
  gcn_epilogue<<<(n4 + 255) / 256, 256, 0, stream>>>(out, bias, pa);
}